// TorchAttention_38757784879418
// MI455X (gfx1250) — compile-verified
//
#include <hip/hip_runtime.h>
#include <math.h>
#include <stdint.h>

#define TB 2
#define TT 2048
#define TD 1024
#define NH 16
#define HD 64

typedef float v2f __attribute__((ext_vector_type(2)));
typedef float v4f __attribute__((ext_vector_type(4)));
typedef float v8f __attribute__((ext_vector_type(8)));
typedef unsigned int u32x4 __attribute__((ext_vector_type(4)));
typedef int i32x4 __attribute__((ext_vector_type(4)));
typedef int i32x8 __attribute__((ext_vector_type(8)));

// Native fp32 WMMA: D(16x16,f32) = A(16x4,f32) * B(4x16,f32) + C
__device__ __forceinline__ v8f wmma_f32x4(v2f a, v2f b, v8f c) {
    return __builtin_amdgcn_wmma_f32_16x16x4_f32(
        /*neg_a=*/false, a, /*neg_b=*/false, b,
        /*c_mod=*/(short)0, c, /*reuse_a=*/false, /*reuse_b=*/false);
}

// ---------------------------------------------------------------------------
// TDM: async 2D-tile DMA Global -> LDS (tensor_load_to_lds, TENSORcnt).
// D# built per CDNA5 ISA §8 (group0/group1 bit layout), 4-byte elements,
// contiguous LDS destination, row stride in elements. Wave-uniform args.
// This toolchain exposes the 6-arg builtin: (g0, g1, g2, g3, g-ext, cpol).
// ---------------------------------------------------------------------------
__device__ __forceinline__ void tdm_load_tile_f32(const float* gptr,
                                                  uint32_t lds_addr,
                                                  int tile_cols, int tile_rows,
                                                  int row_stride_elems) {
    uint64_t ga = (uint64_t)(uintptr_t)gptr;
    u32x4 g0;
    g0[0] = 1u;                                               // count=1, user desc
    g0[1] = lds_addr;                                         // lds_addr [63:32]
    g0[2] = (uint32_t)ga;                                     // global_addr[31:0]
    g0[3] = (uint32_t)((ga >> 32) & 0x01FFFFFFu) | (2u << 30); // ga[56:32] | type=2
    i32x8 g1;
    g1[0] = 2 << 16;                                          // data_size = 4B
    g1[1] = (tile_cols & 0xFFFF) << 16;                       // tensor_dim0[15:0]
    g1[2] = ((unsigned)tile_cols >> 16) | ((tile_rows & 0xFFFF) << 16); // d0 hi | tensor_dim1 lo
    g1[3] = ((unsigned)tile_rows >> 16) | ((tile_cols & 0xFFFF) << 16); // d1 hi | tile_dim0
    g1[4] = tile_rows & 0xFFFF;                               // tile_dim1 (tile_dim2=0)
    g1[5] = row_stride_elems;                                 // tensor_dim0_stride[31:0]
    g1[6] = 0;                                                // stride hi | dim1_stride lo
    g1[7] = 0;
    i32x4 z4 = {0, 0, 0, 0};
    i32x8 z8 = {0, 0, 0, 0, 0, 0, 0, 0};
    __builtin_amdgcn_tensor_load_to_lds(g0, g1, z4, z4, z8, 0);
}

__device__ __forceinline__ uint32_t lds_addr_of(const void* p) {
    // LDS aperture: flat addr low 32 bits are the LDS byte offset.
    return (uint32_t)(uintptr_t)p;
}

// ---------------------------------------------------------------------------
// Tiled GEMM with bias: C[M,N] = A[M,K] @ B[K,N] + bias[N]   (all row-major)
// Block: 256 threads (8 waves). Block tile 64(M) x 128(N), BK=32 LDS staging.
// Waves in 2x4 grid, each wave computes a 32x32 tile = 2x2 WMMA accumulators.
// ---------------------------------------------------------------------------
__global__ __launch_bounds__(256) void gemm_bias_kernel(
    const float* __restrict__ A, const float* __restrict__ Bm,
    const float* __restrict__ bias, float* __restrict__ C,
    int M, int N, int K) {
    __shared__ float sA[64 * 32];    // 8 KB
    __shared__ float sB[32 * 128];   // 16 KB

    const int tid  = threadIdx.x;
    const int lane = tid & 31;
    const int w    = tid >> 5;
    const int half = (lane >> 4) & 1;
    const int l16  = lane & 15;
    const int wm   = (w >> 2) * 32;   // wave row offset within block tile
    const int wn   = (w & 3) * 32;    // wave col offset within block tile

    const int mBase = blockIdx.y * 64;
    const int nBase = blockIdx.x * 128;

    // per-thread staging coordinates
    const int rA = (tid * 8) >> 5, cA = (tid * 8) & 31;
    const int rB = (tid * 16) >> 7, cB = (tid * 16) & 127;

    v8f acc[2][2] = {};

    for (int kb = 0; kb < K; kb += 32) {
        // stage A tile 64x32 (8 floats/thread)
        {
            const float* g = A + (size_t)(mBase + rA) * K + kb + cA;
            *(v4f*)&sA[rA * 32 + cA]     = *(const v4f*)g;
            *(v4f*)&sA[rA * 32 + cA + 4] = *(const v4f*)(g + 4);
        }
        // stage B tile 32x128 (16 floats/thread)
        {
            const float* g = Bm + (size_t)(kb + rB) * N + nBase + cB;
#pragma unroll
            for (int q = 0; q < 4; ++q)
                *(v4f*)&sB[rB * 128 + cB + q * 4] = *(const v4f*)(g + q * 4);
        }
        // prefetch next K-step tiles into cache while this one computes
        if (kb + 32 < K) {
            __builtin_prefetch(A + (size_t)(mBase + rA) * K + kb + 32 + cA, 0, 3);
            __builtin_prefetch(Bm + (size_t)(kb + 32 + rB) * N + nBase + cB, 0, 3);
        }
        __syncthreads();

#pragma unroll
        for (int kk = 0; kk < 32; kk += 4) {
            v2f af[2], bf[2];
#pragma unroll
            for (int mt = 0; mt < 2; ++mt) {
                // A frag: lanes 0-15 -> K = kk,kk+1 ; lanes 16-31 -> kk+2,kk+3
                const float* p = &sA[(wm + mt * 16 + l16) * 32 + kk + 2 * half];
                af[mt].x = p[0];
                af[mt].y = p[1];
            }
#pragma unroll
            for (int nt = 0; nt < 2; ++nt) {
                const float* p = &sB[(size_t)(kk + 2 * half) * 128 + wn + nt * 16 + l16];
                bf[nt].x = p[0];
                bf[nt].y = p[128];
            }
#pragma unroll
            for (int mt = 0; mt < 2; ++mt)
#pragma unroll
                for (int nt = 0; nt < 2; ++nt)
                    acc[mt][nt] = wmma_f32x4(af[mt], bf[nt], acc[mt][nt]);
        }
        __syncthreads();
    }

    // epilogue: bias + store (C layout: VGPR v -> row v / v+8, col = lane&15)
#pragma unroll
    for (int nt = 0; nt < 2; ++nt) {
        int col  = nBase + wn + nt * 16 + l16;
        float bv = bias[col];
#pragma unroll
        for (int mt = 0; mt < 2; ++mt) {
#pragma unroll
            for (int v = 0; v < 8; ++v) {
                int row = mBase + wm + mt * 16 + v + 8 * half;
                C[(size_t)row * N + col] = acc[mt][nt][v] + bv;
            }
        }
    }
}

// ---------------------------------------------------------------------------
// Fused causal flash attention over packed qkv [B, T, 3D].
// Grid: (T/64, B*H). Block: 128 threads (4 waves); each wave owns 16 q-rows.
// K/V tiles (16x64) streamed via TDM tensor_load_to_lds with double buffering:
// wave 0 issues DMA for tile i+1, split-waits TENSORcnt<=2 for tile i, then the
// block barrier releases the WMMA work for tile i while tile i+1 is in flight.
// Output: attn_out [B, T, D] with head h at columns h*HD .. h*HD+63.
// ---------------------------------------------------------------------------
__global__ __launch_bounds__(128) void flash_attn_kernel(
    const float* __restrict__ qkv, float* __restrict__ attn_out) {
    __shared__ float sQ[64 * 64];        // 16 KB, pre-scaled by 1/sqrt(hd)
    __shared__ float sK[2][16 * 64];     // 8 KB, double-buffered TDM dest
    __shared__ float sV[2][16 * 64];     // 8 KB, double-buffered TDM dest
    __shared__ float sP[4][16 * 16];     // per-wave P scratch (C->A relayout)

    const int tid  = threadIdx.x;
    const int lane = tid & 31;
    const int w    = tid >> 5;
    const int half = (lane >> 4) & 1;
    const int l16  = lane & 15;

    const int bh = blockIdx.y;
    const int b  = bh / NH, h = bh % NH;
    const int qTile = blockIdx.x * 64;

    const float* Qb = qkv + (size_t)b * TT * 3 * TD + h * HD;  // row stride 3D
    const float* Kb = Qb + TD;
    const float* Vb = Qb + 2 * TD;

    // stage Q tile 64x64, pre-scaled by softmax scale
    {
        int e = tid * 32;
        int r = e >> 6, c = e & 63;
        const float* g = Qb + (size_t)(qTile + r) * (3 * TD) + c;
#pragma unroll
        for (int q = 0; q < 8; ++q) {
            v4f t = *(const v4f*)(g + q * 4);
            t *= 0.125f;  // 1/sqrt(64)
            *(v4f*)&sQ[r * 64 + c + q * 4] = t;
        }
    }

    float m[8], l[8];
#pragma unroll
    for (int v = 0; v < 8; ++v) { m[v] = -INFINITY; l[v] = 0.f; }
    v8f o[4] = {};

    const int qr0    = qTile + w * 16;
    const int nTiles = (qTile + 64) / 16;

    // prologue: DMA tile 0 into buffer 0
    if (w == 0) {
        tdm_load_tile_f32(Kb, lds_addr_of(&sK[0][0]), HD, 16, 3 * TD);
        tdm_load_tile_f32(Vb, lds_addr_of(&sV[0][0]), HD, 16, 3 * TD);
    }

    for (int it = 0; it < nTiles; ++it) {
        const int buf = it & 1;
        const int jb  = it * 16;

        if (w == 0) {
            if (it + 1 < nTiles) {
                const float* kn = Kb + (size_t)(jb + 16) * (3 * TD);
                const float* vn = Vb + (size_t)(jb + 16) * (3 * TD);
                tdm_load_tile_f32(kn, lds_addr_of(&sK[buf ^ 1][0]), HD, 16, 3 * TD);
                tdm_load_tile_f32(vn, lds_addr_of(&sV[buf ^ 1][0]), HD, 16, 3 * TD);
                __builtin_amdgcn_s_wait_tensorcnt(2);  // tile `it` landed
            } else {
                __builtin_amdgcn_s_wait_tensorcnt(0);
            }
        }
        __syncthreads();  // K/V tile `it` visible; Q staged (first iter)

        // S(16x16) = Q_w(16xhd) @ K^T(hdx16): contraction over hd, 16 WMMA
        v8f s = {};
#pragma unroll
        for (int dd = 0; dd < HD; dd += 4) {
            v2f a, bf;
            const float* pa = &sQ[(w * 16 + l16) * 64 + dd + 2 * half];
            a.x = pa[0]; a.y = pa[1];
            // B(d, n=key) = K[jb+n][d]  -> sK[buf][n*64 + d]
            const float* pb = &sK[buf][l16 * 64 + dd + 2 * half];
            bf.x = pb[0]; bf.y = pb[1];
            s = wmma_f32x4(a, bf, s);
        }

        // causal mask + online softmax update
        const int colk = jb + l16;
#pragma unroll
        for (int v = 0; v < 8; ++v) {
            int rowq = qr0 + v + 8 * half;
            float sv = (colk <= rowq) ? s[v] : -INFINITY;
            float mx = sv;
#pragma unroll
            for (int off = 8; off >= 1; off >>= 1)
                mx = fmaxf(mx, __shfl_xor(mx, off, 16));
            float mn   = fmaxf(m[v], mx);
            float corr = __expf(m[v] - mn);
            float p    = __expf(sv - mn);
            float rs   = p;
#pragma unroll
            for (int off = 8; off >= 1; off >>= 1)
                rs += __shfl_xor(rs, off, 16);
            l[v] = l[v] * corr + rs;
            m[v] = mn;
#pragma unroll
            for (int t = 0; t < 4; ++t) o[t][v] *= corr;
            sP[w][(v + 8 * half) * 16 + l16] = p;  // C-layout -> LDS
        }
        __syncthreads();  // make P visible across lanes

        // O(16x64) += P(16x16) @ V(16x64): 16 WMMA
#pragma unroll
        for (int ct = 0; ct < 4; ++ct) {
#pragma unroll
            for (int kk = 0; kk < 16; kk += 4) {
                v2f a, bf;
                const float* pa = &sP[w][l16 * 16 + kk + 2 * half];
                a.x = pa[0]; a.y = pa[1];
                const float* pb = &sV[buf][(kk + 2 * half) * 64 + ct * 16 + l16];
                bf.x = pb[0]; bf.y = pb[64];
                o[ct] = wmma_f32x4(a, bf, o[ct]);
            }
        }
        __syncthreads();  // all reads of buf done before TDM reuses it
    }

    // normalize and write [B, T, D]
#pragma unroll
    for (int v = 0; v < 8; ++v) {
        int rowq  = qr0 + v + 8 * half;
        float inv = 1.f / l[v];
        float* dst = attn_out + (size_t)(b * TT + rowq) * TD + h * HD;
#pragma unroll
        for (int ct = 0; ct < 4; ++ct)
            dst[ct * 16 + l16] = o[ct][v] * inv;
    }
}

// ---------------------------------------------------------------------------
extern "C" void kernel_launch(void* const* d_in, const int* in_sizes, int n_in,
                              void* d_out, int out_size, void* d_ws, size_t ws_size,
                              hipStream_t stream) {
    (void)in_sizes; (void)n_in; (void)out_size; (void)ws_size;
    const float* x     = (const float*)d_in[0];  // [B,T,D]
    const float* Wqkv  = (const float*)d_in[1];  // [D,3D]
    const float* bqkv  = (const float*)d_in[2];  // [3D]
    const float* Wproj = (const float*)d_in[3];  // [D,D]
    const float* bproj = (const float*)d_in[4];  // [D]
    float* out = (float*)d_out;                  // [B,T,D]

    float* qkv  = (float*)d_ws;                          // [B*T, 3D]
    float* attn = qkv + (size_t)(TB * TT) * (3 * TD);    // [B*T, D]

    const int M = TB * TT;  // 4096

    // 1) qkv = x @ Wqkv + bqkv
    {
        dim3 grid((3 * TD) / 128, M / 64);
        gemm_bias_kernel<<<grid, 256, 0, stream>>>(x, Wqkv, bqkv, qkv, M, 3 * TD, TD);
    }
    // 2) fused causal flash attention (TDM-streamed K/V)
    {
        dim3 grid(TT / 64, TB * NH);
        flash_attn_kernel<<<grid, 128, 0, stream>>>(qkv, attn);
    }
    // 3) out = attn @ Wproj + bproj
    {
        dim3 grid(TD / 128, M / 64);
        gemm_bias_kernel<<<grid, 256, 0, stream>>>(attn, Wproj, bproj, out, M, TD, TD);
    }
}